// LearnedRotation_34952443854911
// MI455X (gfx1250) — compile-verified
//
#include <hip/hip_runtime.h>
#include <math.h>

typedef float v2f __attribute__((ext_vector_type(2)));
typedef float v8f __attribute__((ext_vector_type(8)));

#define IMG_H 512
#define IMG_W 512
#define NCH   16
#define NB    8
#define PIXELS (IMG_H * IMG_W)        // 262144 = 2^18
#define TILES  (PIXELS / 4)           // 65536 4-pixel K-tiles per batch
#define RED_BLOCKS_PER_BATCH 64
#define RED_WAVES_PER_BATCH  (RED_BLOCKS_PER_BATCH * 8)   // 512 (256 thr = 8 waves)
#define PI_F   3.14159265358979323846f

// ---------------------------------------------------------------------------
// Kernel 0: zero the per-(batch,channel) sum accumulator (128 floats in ws).
// ---------------------------------------------------------------------------
__global__ void lr_zero_acc(float* __restrict__ acc) {
    if (threadIdx.x < NB * NCH) acc[threadIdx.x] = 0.0f;
}

// ---------------------------------------------------------------------------
// Kernel 1: channel-sum reduction via V_WMMA_F32_16X16X4_F32.
//   D = ones[16x4] * B[4x16] + C  => every row of D accumulates the per-
//   channel column sums of B. Each WMMA consumes 4 pixels x 16 channels.
//   B layout (f32 4x16, 2 VGPRs): VGPR j, lanes 0-15  -> K=j,   N=lane
//                                  VGPR j, lanes 16-31 -> K=2+j, N=lane-16
//   (any K permutation is fine: A is all-ones so D is K-order invariant;
//    only the N <-> lane&15 mapping matters).
//   Wave id is readfirstlane'd into an SGPR so the trip loop is scalar
//   (s_cmp/s_cbranch) and EXEC stays untouched around the WMMAs.
// ---------------------------------------------------------------------------
__global__ __launch_bounds__(256) void lr_mean_reduce_wmma(
        const float* __restrict__ x, float* __restrict__ acc) {
    const int batch = blockIdx.y;
    const int lane  = threadIdx.x & 31;
    const int wave  = __builtin_amdgcn_readfirstlane(
                          blockIdx.x * (256 >> 5) + (threadIdx.x >> 5));

    const float* __restrict__ xb = x + (size_t)batch * PIXELS * NCH;

    // lane -> element index inside a 64-float (4-pixel) K-tile
    const int e0 = ((lane & 16) << 1) + (lane & 15);   // K slot 0
    const int e1 = e0 + 16;                            // K slot 1

    v2f a = {1.0f, 1.0f};   // all-ones A matrix (16x4)
    v8f c = {};             // f32 accumulator

    #pragma unroll 4
    for (int t = wave; t < TILES; t += RED_WAVES_PER_BATCH) {
        const float* __restrict__ p = xb + (size_t)t * 64;
        v2f bv;
        bv.x = p[e0];
        bv.y = p[e1];
        // 8 args: (neg_a, A, neg_b, B, c_mod, C, reuse_a, reuse_b)
        c = __builtin_amdgcn_wmma_f32_16x16x4_f32(
                false, a, false, bv, (short)0, c, false, false);
    }

    // D VGPR0: lanes 0-15 hold D[m][lane] — all rows equal the column sums.
    if (lane < 16) {
        atomicAdd(&acc[batch * NCH + lane], c[0]);
    }
}

// ---------------------------------------------------------------------------
// Kernel 2: angle head + affine parameter precompute (8 batches, 1 wave).
//   params[b] = { cos, sin, x_off, y_off }
// ---------------------------------------------------------------------------
__global__ void lr_angle(const float* __restrict__ acc,
                         const float* __restrict__ w_angle,
                         const float* __restrict__ b_angle,
                         float* __restrict__ params) {
    const int b = threadIdx.x;
    if (b >= NB) return;
    const float inv_n = 1.0f / (float)PIXELS;
    float dot = b_angle[0];
    #pragma unroll
    for (int ch = 0; ch < NCH; ++ch) {
        dot += (acc[b * NCH + ch] * inv_n) * w_angle[ch];
    }
    float ang = tanhf(dot) * PI_F;
    ang = fminf(fmaxf(ang, -PI_F), PI_F);
    const float cs = cosf(ang);
    const float sn = sinf(ang);
    const float wm1 = (float)(IMG_W - 1);
    const float hm1 = (float)(IMG_H - 1);
    const float xo = (wm1 - (cs * wm1 - sn * hm1)) * 0.5f;
    const float yo = (hm1 - (sn * wm1 + cs * hm1)) * 0.5f;
    params[b * 4 + 0] = cs;
    params[b * 4 + 1] = sn;
    params[b * 4 + 2] = xo;
    params[b * 4 + 3] = yo;
}

// ---------------------------------------------------------------------------
// Kernel 3: bilinear rotation, one thread per output pixel (16 channels).
//   4 taps x 4 float4 loads (64B contiguous per tap), 4 float4 stores.
//   OOB taps: weight forced to 0 (matches reference read() * valid).
// ---------------------------------------------------------------------------
__global__ __launch_bounds__(256) void lr_rotate(
        const float* __restrict__ x, const float* __restrict__ params,
        float* __restrict__ out) {
    const int gid = blockIdx.x * blockDim.x + threadIdx.x;
    const int b   = gid >> 18;              // PIXELS = 2^18
    const int rem = gid & (PIXELS - 1);
    const int oy  = rem >> 9;               // IMG_W = 2^9
    const int ox  = rem & (IMG_W - 1);

    const float cs = params[b * 4 + 0];
    const float sn = params[b * 4 + 1];
    const float xo = params[b * 4 + 2];
    const float yo = params[b * 4 + 3];

    const float fox = (float)ox, foy = (float)oy;
    const float ix = cs * fox - sn * foy + xo;
    const float iy = sn * fox + cs * foy + yo;
    const float x0 = floorf(ix), y0 = floorf(iy);
    const float wx = ix - x0,    wy = iy - y0;

    const float* __restrict__ img = x + (size_t)b * PIXELS * NCH;
    const float wm1 = (float)(IMG_W - 1);
    const float hm1 = (float)(IMG_H - 1);

    float a0x=0,a0y=0,a0z=0,a0w=0, a1x=0,a1y=0,a1z=0,a1w=0;
    float a2x=0,a2y=0,a2z=0,a2w=0, a3x=0,a3y=0,a3z=0,a3w=0;

    #pragma unroll
    for (int ty = 0; ty < 2; ++ty) {
        const float yy  = y0 + (float)ty;
        const float wyv = ty ? wy : (1.0f - wy);
        const bool  vy  = (yy >= 0.0f) && (yy <= hm1);
        const int   yi  = (int)fminf(fmaxf(yy, 0.0f), hm1);
        #pragma unroll
        for (int tx = 0; tx < 2; ++tx) {
            const float xx  = x0 + (float)tx;
            const float wxv = tx ? wx : (1.0f - wx);
            const bool  vx  = (xx >= 0.0f) && (xx <= wm1);
            const int   xi  = (int)fminf(fmaxf(xx, 0.0f), wm1);
            const float wgt = (vx && vy) ? (wxv * wyv) : 0.0f;

            const float4* __restrict__ p4 =
                (const float4*)(img + ((size_t)yi * IMG_W + xi) * NCH);
            const float4 t0 = p4[0], t1 = p4[1], t2 = p4[2], t3 = p4[3];
            a0x += wgt * t0.x; a0y += wgt * t0.y; a0z += wgt * t0.z; a0w += wgt * t0.w;
            a1x += wgt * t1.x; a1y += wgt * t1.y; a1z += wgt * t1.z; a1w += wgt * t1.w;
            a2x += wgt * t2.x; a2y += wgt * t2.y; a2z += wgt * t2.z; a2w += wgt * t2.w;
            a3x += wgt * t3.x; a3y += wgt * t3.y; a3z += wgt * t3.z; a3w += wgt * t3.w;
        }
    }

    float4* __restrict__ o4 = (float4*)(out + (size_t)gid * NCH);
    o4[0] = make_float4(a0x, a0y, a0z, a0w);
    o4[1] = make_float4(a1x, a1y, a1z, a1w);
    o4[2] = make_float4(a2x, a2y, a2z, a2w);
    o4[3] = make_float4(a3x, a3y, a3z, a3w);
}

// ---------------------------------------------------------------------------
extern "C" void kernel_launch(void* const* d_in, const int* in_sizes, int n_in,
                              void* d_out, int out_size, void* d_ws, size_t ws_size,
                              hipStream_t stream) {
    const float* x       = (const float*)d_in[0];   // [8,512,512,16] f32
    const float* w_angle = (const float*)d_in[1];   // [16,1] f32
    const float* b_angle = (const float*)d_in[2];   // [1] f32
    float* out = (float*)d_out;

    float* acc    = (float*)d_ws;   // 128 floats: per-(batch,channel) sums
    float* params = acc + 128;      // 32 floats: {cos,sin,xoff,yoff} per batch

    lr_zero_acc<<<1, 128, 0, stream>>>(acc);

    dim3 rgrid(RED_BLOCKS_PER_BATCH, NB);   // 512 waves per batch
    lr_mean_reduce_wmma<<<rgrid, 256, 0, stream>>>(x, acc);

    lr_angle<<<1, 32, 0, stream>>>(acc, w_angle, b_angle, params);

    lr_rotate<<<(NB * PIXELS) / 256, 256, 0, stream>>>(x, params, out);
}